// GNNEmbeddings_8581344658127
// MI455X (gfx1250) — compile-verified
//
#include <hip/hip_runtime.h>
#include <math.h>

typedef __attribute__((ext_vector_type(2))) float v2f;
typedef __attribute__((ext_vector_type(8))) float v8f;

#define NN     102   // nodes
#define MPAD   112   // M padded to 7 x 16
#define APITCH 68    // act row pitch in floats (68%64==4 -> conflict-free ds_load_b64)

// ---- compile-time graph (CSR with self-loops, in-degree based norm) ----
constexpr int NE_RAW = 84;
constexpr int E_RAW[NE_RAW][2] = {
 {0,6},{0,5},{6,8},{5,7},{0,62},{62,63},{63,64},{59,64},{59,60},{60,61},{61,62},
 {0,74},{71,72},{72,73},{73,74},{74,75},{75,76},{76,77},{77,78},{78,79},{79,80},
 {80,81},{81,82},{71,82},{71,83},{77,87},{83,84},{84,85},{85,86},{86,87},{87,88},
 {88,89},{89,90},{83,90},{0,65},{65,66},{66,67},{67,68},{68,69},{69,70},{65,70},
 {7,91},{91,92},{92,93},{93,94},{94,95},{91,96},{96,97},{97,98},{98,99},{91,100},
 {100,101},{101,102},{102,103},{91,104},{104,105},{105,106},{106,107},{91,108},
 {108,109},{109,110},{110,111},{8,112},{112,113},{113,114},{114,115},{115,116},
 {112,117},{117,118},{118,119},{119,120},{112,121},{121,122},{122,123},{123,124},
 {112,125},{125,126},{126,127},{127,128},{112,129},{129,130},{130,131},{131,132}};

struct Graph {
  int off[NN + 1];
  int row[NN + NE_RAW];   // 155 used
};

constexpr Graph build_graph() {
  Graph g{};
  int kr[NE_RAW] = {}, kc[NE_RAW] = {};
  int ne = 0;
  for (int e = 0; e < NE_RAW; ++e)
    if (E_RAW[e][0] < NN && E_RAW[e][1] < NN) { kr[ne] = E_RAW[e][0]; kc[ne] = E_RAW[e][1]; ++ne; }
  int deg[NN] = {};
  for (int n = 0; n < NN; ++n) deg[n] = 1;          // self loop
  for (int e = 0; e < ne; ++e) deg[kc[e]]++;
  g.off[0] = 0;
  for (int n = 0; n < NN; ++n) g.off[n + 1] = g.off[n] + deg[n];
  int fill[NN] = {};
  for (int n = 0; n < NN; ++n) { g.row[g.off[n]] = n; fill[n] = 1; }
  for (int e = 0; e < ne; ++e) { int c = kc[e]; g.row[g.off[c] + fill[c]] = kr[e]; fill[c]++; }
  return g;
}

__constant__ Graph g_G = build_graph();

// ---- fused GCN + projection kernel ----
__global__ __launch_bounds__(256) void gnn_fused_kernel(
    const float* __restrict__ x,    // (B, 204)
    const float* __restrict__ Wg,   // (2, 64)
    const float* __restrict__ bg,   // (64)
    const float* __restrict__ Wp,   // (64, 128)
    const float* __restrict__ bp,   // (128)
    float* __restrict__ out,        // (B, 102, 128)
    int batch_per_block, int nbatch)
{
  __shared__ float lds_act[MPAD * APITCH];   // padded act matrix (M=112, K=64)
  __shared__ float lds_x[2 * NN];
  __shared__ float lds_y[2 * NN];
  __shared__ float lds_wg[2 * 64];
  __shared__ float lds_bg[64];

  const int t    = threadIdx.x;
  const int lane = t & 31;
  const int wv   = t >> 5;        // wave 0..7 owns N-tile wv (cols 16*wv..16*wv+15)
  const int m    = lane & 15;
  const int hi   = lane >> 4;     // lane half selects K sub-pair in WMMA frags

  // Preload B-fragments (W_proj 64x16 slab) for this wave's N-tile: kept in VGPRs.
  // B frag layout (16x16x4 f32): lanes 0-15 N=m K={k0,k0+1}; lanes 16-31 N=m K={k0+2,k0+3}
  v2f bfr[16];
  const int col = wv * 16 + m;
  #pragma unroll
  for (int ks = 0; ks < 16; ++ks) {
    const int k = ks * 4 + hi * 2;
    bfr[ks].x = Wp[k * 128 + col];
    bfr[ks].y = Wp[(k + 1) * 128 + col];
  }
  const float bias_p = bp[col];

  // Zero act (rows 102..111 stay zero forever); stage W_gcn / b_gcn
  for (int q = t; q < MPAD * APITCH; q += 256) lds_act[q] = 0.0f;
  if (t < 64) { lds_wg[t] = Wg[t]; lds_wg[64 + t] = Wg[64 + t]; lds_bg[t] = bg[t]; }
  __syncthreads();

  for (int it = 0; it < batch_per_block; ++it) {
    const int bi = blockIdx.x * batch_per_block + it;
    if (bi >= nbatch) break;

    // stage x[bi] (102 nodes x 2 features)
    for (int q = t; q < 2 * NN; q += 256) lds_x[q] = x[(size_t)bi * (2 * NN) + q];
    __syncthreads();

    // aggregate in 2-dim feature space: y[n] = sum_{r in N(n)} norm(r,n) * x2[r]
    if (t < NN) {
      const int o0 = g_G.off[t], o1 = g_G.off[t + 1];
      const float dn = 1.0f / sqrtf((float)(o1 - o0));   // dinv[col]
      float y0 = 0.f, y1 = 0.f;
      for (int j = o0; j < o1; ++j) {
        const int r = g_G.row[j];
        const float dr = (float)(g_G.off[r + 1] - g_G.off[r]);
        const float w = dn * (1.0f / sqrtf(dr));          // dinv[row]*dinv[col]
        y0 += w * lds_x[2 * r];
        y1 += w * lds_x[2 * r + 1];
      }
      lds_y[2 * t] = y0; lds_y[2 * t + 1] = y1;
    }
    __syncthreads();

    // act[n][h] = relu(y0*Wg[0][h] + y1*Wg[1][h] + bg[h])
    for (int e = t; e < NN * 64; e += 256) {
      const int n = e >> 6, h = e & 63;
      float v = lds_y[2 * n] * lds_wg[h] + lds_y[2 * n + 1] * lds_wg[64 + h] + lds_bg[h];
      lds_act[n * APITCH + h] = v > 0.f ? v : 0.f;
    }
    __syncthreads();

    // projection: out(112x128) = act(112x64) @ Wp(64x128), per wave: 7 M-tiles,
    // K=64 as 16 chained v_wmma_f32_16x16x4_f32
    for (int i = 0; i < 7; ++i) {
      v8f acc = {};
      // A frag: lanes 0-15 M=m K={k0,k0+1}; lanes 16-31 M=m K={k0+2,k0+3}
      const float* arow = &lds_act[(i * 16 + m) * APITCH + hi * 2];
      #pragma unroll
      for (int ks = 0; ks < 16; ++ks) {
        v2f a = *(const v2f*)(arow + ks * 4);
        acc = __builtin_amdgcn_wmma_f32_16x16x4_f32(
            false, a, false, bfr[ks], (short)0, acc, false, false);
      }
      // C/D layout: VGPR r -> row (i*16 + r + hi*8), col = 16*wv + m
      #pragma unroll
      for (int r = 0; r < 8; ++r) {
        const int row = i * 16 + hi * 8 + r;
        if (row < NN)
          out[((size_t)bi * NN + row) * 128 + col] = acc[r] + bias_p;
      }
    }
    __syncthreads();
  }
}

extern "C" void kernel_launch(void* const* d_in, const int* in_sizes, int n_in,
                              void* d_out, int out_size, void* d_ws, size_t ws_size,
                              hipStream_t stream) {
  const float* x  = (const float*)d_in[0];
  const float* Wg = (const float*)d_in[1];
  const float* bg = (const float*)d_in[2];
  const float* Wp = (const float*)d_in[3];
  const float* bp = (const float*)d_in[4];
  float* out = (float*)d_out;

  const int B   = in_sizes[0] / (2 * NN);   // 8192
  const int per = 8;                        // batch elements per workgroup
  const int blocks = (B + per - 1) / per;   // 1024
  gnn_fused_kernel<<<blocks, 256, 0, stream>>>(x, Wg, bg, Wp, bp, out, per, B);
}